// SpLayer_57707180589153
// MI455X (gfx1250) — compile-verified
//
#include <hip/hip_runtime.h>
#include <hip/hip_bf16.h>

#ifndef __has_builtin
#define __has_builtin(x) 0
#endif

#define TPB 256
#define PTS_PER_THREAD 4
#define PTS_PER_BLOCK (TPB * PTS_PER_THREAD)   // 1024 points / block
#define TILE_FLOATS (PTS_PER_BLOCK * 3)        // 3072 floats = 12288 B
#define CHUNK_BYTES (TPB * 16)                 // 4096 B per async sweep

#define FACTOR_DEG 57.29577951308232f          // 180/pi
#define PI_F       3.14159265358979323846f
#define PIO2_F     1.5707963267948966f

typedef int v4i __attribute__((ext_vector_type(4)));
typedef __attribute__((address_space(1))) v4i glob_v4i;
typedef __attribute__((address_space(3))) v4i lds_v4i;
typedef __attribute__((address_space(3))) void lds_void;
typedef __attribute__((address_space(1))) void glob_void;

// ---- CDNA5 async global -> LDS copy (ASYNCcnt path) ------------------------
__device__ __forceinline__ void async_copy_b128(const void* g, void* l) {
#if __has_builtin(__builtin_amdgcn_global_load_async_to_lds_b128)
  __builtin_amdgcn_global_load_async_to_lds_b128(
      (glob_v4i*)g, (lds_v4i*)l, /*offset=*/0, /*cpol=*/0);
#else
  unsigned lds_off = (unsigned)(unsigned long long)(lds_void*)l;
  unsigned long long ga = (unsigned long long)(glob_void*)g;
  asm volatile("global_load_async_to_lds_b128 %0, %1, off"
               :: "v"(lds_off), "v"(ga) : "memory");
#endif
}

__device__ __forceinline__ void wait_async0() {
#if __has_builtin(__builtin_amdgcn_s_wait_asynccnt)
  __builtin_amdgcn_s_wait_asynccnt(0);
#else
  asm volatile("s_wait_asynccnt 0" ::: "memory");
#endif
}

// ---- branchless single-precision acos (Cephes-style minimax) ---------------
__device__ __forceinline__ float fast_acosf(float x) {
  float a = __builtin_fabsf(x);
  bool big = a > 0.5f;
  float t = big ? __builtin_fmaf(-0.5f, a, 0.5f) : a * a;       // z = s^2
  float s = big ? __builtin_amdgcn_sqrtf(t) : a;                // s in [0, .707]
  float p = __builtin_fmaf(t, 4.2163199048e-2f, 2.4181311049e-2f);
  p = __builtin_fmaf(t, p, 4.5470025998e-2f);
  p = __builtin_fmaf(t, p, 7.4953002686e-2f);
  p = __builtin_fmaf(t, p, 1.6666752422e-1f);
  float r = __builtin_fmaf(s * t, p, s);                        // asin(s)
  float pos = big ? (2.0f * r) : (PIO2_F - r);                  // acos(|x|)
  return (x < 0.0f) ? (PI_F - pos) : pos;
}

// ---- per-point spherical-angle math ----------------------------------------
__device__ __forceinline__ float2 sph_compute(float xx, float yy, float zz) {
  float nxy2 = __builtin_fmaf(xx, xx, yy * yy);
  float r2   = __builtin_fmaf(zz, zz, nxy2);
  float cph  = zz * __builtin_amdgcn_rsqf(r2);    // z / |v|
  float cps  = xx * __builtin_amdgcn_rsqf(nxy2);  // x / |v_xy|
  cph = fminf(fmaxf(cph, -1.0f), 1.0f);
  cps = fminf(fmaxf(cps, -1.0f), 1.0f);
  float phi = fast_acosf(cph);
  float psi = fast_acosf(cps);
  psi = (yy < 0.0f) ? -psi : psi;                 // sign from y
  phi = -__builtin_fmaf(phi, FACTOR_DEG, 15.0f);  // LONLY: phi = -(deg + 15)
  psi = psi * FACTOR_DEG;
  return make_float2(phi, psi);
}

__global__ __launch_bounds__(TPB) void sph_angles_kernel(
    const float* __restrict__ x, float2* __restrict__ out, long long nPts) {
  __shared__ float tile[TILE_FLOATS];
  const int t = threadIdx.x;
  const long long blockPt = (long long)blockIdx.x * PTS_PER_BLOCK;

  if (blockPt + PTS_PER_BLOCK <= nPts) {
    // Fast path: DMA the whole 12 KB AoS tile into LDS, fully coalesced.
    const char* gbase = (const char*)x + blockPt * 12ll;
    char* lbase = (char*)tile;
#pragma unroll
    for (int k = 0; k < 3; ++k) {
      int off = k * CHUNK_BYTES + t * 16;
      async_copy_b128(gbase + off, lbase + off);
    }
    wait_async0();
    __syncthreads();

#pragma unroll
    for (int j = 0; j < PTS_PER_THREAD; ++j) {
      int p = t + j * TPB;   // lane stride 12 B in LDS -> conflict-free
      float xx = tile[3 * p + 0];
      float yy = tile[3 * p + 1];
      float zz = tile[3 * p + 2];
      out[blockPt + p] = sph_compute(xx, yy, zz);
    }
  } else {
    // Tail path (not hit for the 64x262144 shape; kept for generality).
    for (int j = 0; j < PTS_PER_THREAD; ++j) {
      long long idx = blockPt + t + (long long)j * TPB;
      if (idx < nPts) {
        float xx = x[3 * idx + 0];
        float yy = x[3 * idx + 1];
        float zz = x[3 * idx + 2];
        out[idx] = sph_compute(xx, yy, zz);
      }
    }
  }
}

extern "C" void kernel_launch(void* const* d_in, const int* in_sizes, int n_in,
                              void* d_out, int out_size, void* d_ws, size_t ws_size,
                              hipStream_t stream) {
  const float* x = (const float*)d_in[0];
  float2* out = (float2*)d_out;
  long long nPts = (long long)in_sizes[0] / 3;   // 16,777,216 for the ref shape
  int blocks = (int)((nPts + PTS_PER_BLOCK - 1) / PTS_PER_BLOCK);
  sph_angles_kernel<<<blocks, TPB, 0, stream>>>(x, out, nPts);
}